// IIR_32615981646404
// MI455X (gfx1250) — compile-verified
//
#include <hip/hip_runtime.h>
#include <hip/hip_bf16.h>

// Problem constants (from reference)
#define SEQ   524288
#define BATCH 16
#define CD    16
#define FL    1024      // FFT length
#define NC    2047      // n_coeffs
#define BS    512       // block size (hop)
#define NB    1024      // number of blocks = SEQ/BS

typedef __attribute__((ext_vector_type(2))) float v2f;
typedef __attribute__((ext_vector_type(8))) float v8f;

// Twiddle table: twr/twi[j] = exp(-2*pi*i*j/1024), j in [0,768)  (serves w, w^2, w^3)
__device__ __forceinline__ void fill_twiddles(float* twr, float* twi, int tid) {
    for (int j = tid; j < 768; j += 256) {
        const float ang = -6.28318530717958647692f * (float)j * (1.0f / 1024.0f);
        twr[j] = __cosf(ang);
        twi[j] = __sinf(ang);
    }
}

// ---------------------------------------------------------------------------
// Radix-4 DIF forward 1024-pt FFT: natural-order input -> digit-reversed output.
// 256 threads, exactly one radix-4 butterfly per thread per stage, 5 stages.
// ---------------------------------------------------------------------------
__device__ __forceinline__ void fft1024_dif_fwd(float* re, float* im,
                                                const float* twr, const float* twi,
                                                int tid) {
    #pragma unroll
    for (int s = 4; s >= 0; --s) {
        const int mh  = 1 << (2 * s);
        const int grp = tid >> (2 * s);
        const int pos = tid & (mh - 1);
        const int i0  = (grp << (2 * s + 2)) + pos;
        const int i1 = i0 + mh, i2 = i0 + 2 * mh, i3 = i0 + 3 * mh;

        const float ar = re[i0], ai = im[i0];
        const float br = re[i1], bi = im[i1];
        const float cr = re[i2], ci = im[i2];
        const float dr = re[i3], di = im[i3];

        const float t0r = ar + cr, t0i = ai + ci;
        const float t1r = ar - cr, t1i = ai - ci;
        const float t2r = br + dr, t2i = bi + di;
        const float t3r = br - dr, t3i = bi - di;

        // forward: -j * t3 = (t3i, -t3r)
        const float u1r = t1r + t3i, u1i = t1i - t3r;   // t1 - j*t3
        const float u3r = t1r - t3i, u3i = t1i + t3r;   // t1 + j*t3
        const float y2r = t0r - t2r, y2i = t0i - t2i;

        const int ti1 = pos << (8 - 2 * s);
        const float w1r = twr[ti1],     w1i = twi[ti1];
        const float w2r = twr[2 * ti1], w2i = twi[2 * ti1];
        const float w3r = twr[3 * ti1], w3i = twi[3 * ti1];

        re[i0] = t0r + t2r;               im[i0] = t0i + t2i;
        re[i1] = u1r * w1r - u1i * w1i;   im[i1] = u1r * w1i + u1i * w1r;
        re[i2] = y2r * w2r - y2i * w2i;   im[i2] = y2r * w2i + y2i * w2r;
        re[i3] = u3r * w3r - u3i * w3i;   im[i3] = u3r * w3i + u3i * w3r;
        __syncthreads();
    }
}

// ---------------------------------------------------------------------------
// Radix-4 DIT inverse 1024-pt FFT: digit-reversed input -> natural-order output.
// Conjugate twiddles; caller applies the 1/N scale at store time.
// ---------------------------------------------------------------------------
__device__ __forceinline__ void fft1024_dit_inv(float* re, float* im,
                                                const float* twr, const float* twi,
                                                int tid) {
    #pragma unroll
    for (int s = 0; s <= 4; ++s) {
        const int mh  = 1 << (2 * s);
        const int grp = tid >> (2 * s);
        const int pos = tid & (mh - 1);
        const int i0  = (grp << (2 * s + 2)) + pos;
        const int i1 = i0 + mh, i2 = i0 + 2 * mh, i3 = i0 + 3 * mh;

        const int ti1 = pos << (8 - 2 * s);
        // conj(table) = inverse twiddles
        const float w1r = twr[ti1],     w1i = -twi[ti1];
        const float w2r = twr[2 * ti1], w2i = -twi[2 * ti1];
        const float w3r = twr[3 * ti1], w3i = -twi[3 * ti1];

        const float ar = re[i0], ai = im[i0];
        const float xr1 = re[i1], xi1 = im[i1];
        const float xr2 = re[i2], xi2 = im[i2];
        const float xr3 = re[i3], xi3 = im[i3];

        const float br = xr1 * w1r - xi1 * w1i, bi = xr1 * w1i + xi1 * w1r;
        const float cr = xr2 * w2r - xi2 * w2i, ci = xr2 * w2i + xi2 * w2r;
        const float dr = xr3 * w3r - xi3 * w3i, di = xr3 * w3i + xi3 * w3r;

        const float t0r = ar + cr, t0i = ai + ci;
        const float t1r = ar - cr, t1i = ai - ci;
        const float t2r = br + dr, t2i = bi + di;
        const float t3r = br - dr, t3i = bi - di;

        re[i0] = t0r + t2r;  im[i0] = t0i + t2i;
        re[i2] = t0r - t2r;  im[i2] = t0i - t2i;
        // inverse: +j * t3 = (-t3i, t3r)
        re[i1] = t1r - t3i;  im[i1] = t1i + t3r;   // t1 + j*t3
        re[i3] = t1r + t3i;  im[i3] = t1i - t3r;   // t1 - j*t3
        __syncthreads();
    }
}

// ---------------------------------------------------------------------------
// Kernel A: filt[16][4094] = c[16][16] @ W^T + bias  via V_WMMA_F32_16X16X4_F32
// One wave per 16-column N-tile; 4 K-steps of 16x4 * 4x16.
// ---------------------------------------------------------------------------
__global__ __launch_bounds__(32) void filt_wmma_kernel(
        const float* __restrict__ cmat, const float* __restrict__ W,
        const float* __restrict__ bias, float* __restrict__ filt) {
    const int tile = blockIdx.x;            // 0..255
    const int lane = threadIdx.x;           // 0..31
    const int half = lane >> 4;
    const int lid  = lane & 15;
    const int n    = tile * 16 + lid;       // output coefficient column
    const int nw   = (n < 2 * NC) ? n : 0;  // safe load index (keeps EXEC uniform)

    v8f acc = {};
    #pragma unroll
    for (int kk = 0; kk < CD; kk += 4) {
        const int k0 = kk + (half ? 2 : 0);
        v2f a, b;
        a.x = cmat[lid * CD + k0];
        a.y = cmat[lid * CD + k0 + 1];
        b.x = W[nw * CD + k0];
        b.y = W[nw * CD + k0 + 1];
        acc = __builtin_amdgcn_wmma_f32_16x16x4_f32(
            /*neg_a=*/false, a, /*neg_b=*/false, b,
            /*c_mod=*/(short)0, acc, /*reuse_a=*/false, /*reuse_b=*/false);
    }

    if (n < 2 * NC) {
        const float bv = bias[n];
        #pragma unroll
        for (int r = 0; r < 8; ++r) {
            const int m = r + half * 8;     // batch row
            filt[m * 4096 + n] = acc[r] + bv;
        }
    }
}

// ---------------------------------------------------------------------------
// Kernel B: per-batch transfer function, in the DIF-scrambled domain:
//   Hs = DIF(b_coeffs) / (DIF(a_coeffs) + 1e-8)
// Consumed only elementwise by kernel D, so the scrambled order is consistent.
// ---------------------------------------------------------------------------
__global__ __launch_bounds__(256) void hfilter_kernel(
        const float* __restrict__ filt, float* __restrict__ Hbuf) {
    __shared__ float bre[FL], bim[FL], are_[FL], aim_[FL];
    __shared__ float twr[768], twi[768];
    const int b   = blockIdx.x;
    const int tid = threadIdx.x;
    const float* f = filt + b * 4096;

    fill_twiddles(twr, twi, tid);
    for (int p = tid; p < FL; p += 256) {     // natural order (coalesced)
        bre[p] = f[p];
        bim[p] = f[NC + p];
        if (p == 0) { are_[p] = 1.0f; aim_[p] = 0.0f; }
        else        { are_[p] = f[FL - 1 + p]; aim_[p] = f[NC + FL - 1 + p]; }
    }
    __syncthreads();

    fft1024_dif_fwd(bre, bim, twr, twi, tid);
    fft1024_dif_fwd(are_, aim_, twr, twi, tid);

    float2* H2 = (float2*)Hbuf;
    for (int p = tid; p < FL; p += 256) {
        const float dr = are_[p] + 1e-8f;
        const float di = aim_[p];
        const float inv = 1.0f / (dr * dr + di * di);
        float2 h;
        h.x = (bre[p] * dr + bim[p] * di) * inv;
        h.y = (bim[p] * dr - bre[p] * di) * inv;
        H2[b * FL + p] = h;
    }
}

// ---------------------------------------------------------------------------
// Kernel C: forward DIF FFTs of input blocks (batch-independent: done ONCE).
// Natural-order coalesced loads; output left in scrambled order.
// ---------------------------------------------------------------------------
__global__ __launch_bounds__(256) void block_fft_kernel(
        const float* __restrict__ x, float* __restrict__ Xbuf) {
    __shared__ float re[FL], im[FL];
    __shared__ float twr[768], twi[768];
    const int k   = blockIdx.x;
    const int tid = threadIdx.x;

    fill_twiddles(twr, twi, tid);
    for (int p = tid; p < FL; p += 256) {
        float v = 0.0f;
        if (p < BS) {
            const int t = k * BS + p;             // index into xp = [1023 zeros, x]
            if (t >= FL - 1) v = x[t - (FL - 1)];
        }
        re[p] = v; im[p] = 0.0f;
    }
    __syncthreads();

    fft1024_dif_fwd(re, im, twr, twi, tid);

    float2* X2 = (float2*)Xbuf;
    for (int p = tid; p < FL; p += 256) {
        float2 v; v.x = re[p]; v.y = im[p];
        X2[k * FL + p] = v;
    }
}

// ---------------------------------------------------------------------------
// Kernel D: Yb[b][k] = real(IDFT(X[k] .* H[b])). One workgroup per (b,k).
// Scrambled-domain elementwise multiply (coalesced float2 loads, X/H L2-hot),
// then radix-4 DIT inverse -> natural order coalesced stores.
// ---------------------------------------------------------------------------
__global__ __launch_bounds__(256) void apply_ifft_kernel(
        const float* __restrict__ Xbuf, const float* __restrict__ Hbuf,
        float* __restrict__ Ybuf) {
    __shared__ float re[FL], im[FL];
    __shared__ float twr[768], twi[768];
    const int wg  = blockIdx.x;
    const int k   = wg & (NB - 1);
    const int b   = wg >> 10;
    const int tid = threadIdx.x;

    fill_twiddles(twr, twi, tid);
    const float2* X2 = (const float2*)Xbuf;
    const float2* H2 = (const float2*)Hbuf;
    for (int p = tid; p < FL; p += 256) {
        const float2 xv = X2[k * FL + p];
        const float2 hv = H2[b * FL + p];
        re[p] = xv.x * hv.x - xv.y * hv.y;
        im[p] = xv.x * hv.y + xv.y * hv.x;
    }
    __syncthreads();

    fft1024_dit_inv(re, im, twr, twi, tid);

    const float scale = 1.0f / (float)FL;
    for (int p = tid; p < FL; p += 256) {
        Ybuf[(size_t)(b * NB + k) * FL + p] = re[p] * scale;
    }
}

// ---------------------------------------------------------------------------
// Kernel E: overlap-add gather, float4-vectorized (branch uniform per group).
// out[b][t] = Y[b][t/512][t%512] + (t>=512 ? Y[b][t/512-1][t%512+512] : 0)
// ---------------------------------------------------------------------------
__global__ __launch_bounds__(256) void overlap_add_kernel(
        const float* __restrict__ Ybuf, float* __restrict__ out) {
    const size_t g = (size_t)blockIdx.x * blockDim.x + threadIdx.x;   // float4 group
    if (g >= (size_t)BATCH * SEQ / 4) return;
    const int b  = (int)(g / (SEQ / 4));
    const int t0 = (int)(g % (SEQ / 4)) * 4;
    const int k  = t0 >> 9;
    const int r  = t0 & (BS - 1);

    const float4* Y4a = (const float4*)(Ybuf + (size_t)(b * NB + k) * FL + r);
    float4 v = *Y4a;
    if (t0 >= BS) {
        const float4* Y4b = (const float4*)(Ybuf + (size_t)(b * NB + (k - 1)) * FL + r + BS);
        const float4 w = *Y4b;
        v.x += w.x; v.y += w.y; v.z += w.z; v.w += w.w;
    }
    *(float4*)(out + (size_t)b * SEQ + t0) = v;
}

// ---------------------------------------------------------------------------
extern "C" void kernel_launch(void* const* d_in, const int* in_sizes, int n_in,
                              void* d_out, int out_size, void* d_ws, size_t ws_size,
                              hipStream_t stream) {
    const float* x    = (const float*)d_in[0];   // (1, 524288, 1)
    const float* cmat = (const float*)d_in[1];   // (16, 1, 16)
    const float* W    = (const float*)d_in[2];   // (4094, 16)
    const float* bias = (const float*)d_in[3];   // (4094,)
    float* out = (float*)d_out;                  // (16, 524288, 1)

    float* ws   = (float*)d_ws;
    float* filt = ws;                            // 16*4096            =    65536 f
    float* Hbuf = filt + 16 * 4096;              // 16*1024*2          =    32768 f
    float* Xbuf = Hbuf + BATCH * FL * 2;         // 1024*1024*2        =  2097152 f
    float* Ybuf = Xbuf + (size_t)NB * FL * 2;    // 16*1024*1024       = 16777216 f

    // A: conditioning GEMM on WMMA (fp32 16x16x4), 256 N-tiles, 1 wave each
    filt_wmma_kernel<<<256, 32, 0, stream>>>(cmat, W, bias, filt);
    // B: per-batch transfer function H (DIF-scrambled domain)
    hfilter_kernel<<<BATCH, 256, 0, stream>>>(filt, Hbuf);
    // C: forward block DIF-FFTs (batch-independent, done once)
    block_fft_kernel<<<NB, 256, 0, stream>>>(x, Xbuf);
    // D: scrambled-domain multiply + radix-4 DIT inverse per (batch, block)
    apply_ifft_kernel<<<BATCH * NB, 256, 0, stream>>>(Xbuf, Hbuf, Ybuf);
    // E: overlap-add (float4)
    const size_t groups = (size_t)BATCH * SEQ / 4;
    overlap_add_kernel<<<(unsigned)((groups + 255) / 256), 256, 0, stream>>>(Ybuf, out);
}